// GradientLoss_84404697301080
// MI455X (gfx1250) — compile-verified
//
#include <hip/hip_runtime.h>
#include <hip/hip_bf16.h>
#include <stdint.h>

#define W      512
#define H      512
#define TH     8            // rows per block band
#define HALO   (TH + 2)     // staged rows incl. top/bottom halo
#define BLOCK  256          // 8 wave32s
#define EPSG   1e-6f

typedef __attribute__((ext_vector_type(2))) float v2f;
typedef __attribute__((ext_vector_type(8))) float v8f;
typedef __attribute__((ext_vector_type(4))) int   v4i;

typedef __attribute__((address_space(1))) v4i g_v4i;   // global int4
typedef __attribute__((address_space(3))) v4i l_v4i;   // LDS int4

// ---------------------------------------------------------------------------
// Stage HALO x 512 fp32 rows of one plane into LDS.
// Uses CDNA5 async global->LDS B128 copies when the builtin exists; rows
// outside the image are zero-filled (SAME zero padding).
// ---------------------------------------------------------------------------
__device__ __forceinline__ void stage_tile(const float* __restrict__ src,
                                           float* sdst, int r0, int tid) {
#pragma unroll
  for (int i = 0; i < (HALO * (W / 4)) / BLOCK; ++i) {   // 10*128/256 = 5
    int g   = i * BLOCK + tid;
    int row = g >> 7;                // g / 128
    int c4  = (g & 127) << 2;        // 4-float column start
    int gr  = r0 - 1 + row;          // global row (can be -1 or H)
    int li  = row * W + c4;          // lds float index
    if (gr >= 0 && gr < H) {
#if __has_builtin(__builtin_amdgcn_global_load_async_to_lds_b128)
      g_v4i* gp = (g_v4i*)(uintptr_t)(src + (size_t)gr * W + c4);
      l_v4i* lp = (l_v4i*)(uint32_t)(uintptr_t)(sdst + li);
      __builtin_amdgcn_global_load_async_to_lds_b128(gp, lp, 0, 0);
#else
      *(float4*)(sdst + li) = *(const float4*)(src + (size_t)gr * W + c4);
#endif
    } else {
      *(float4*)(sdst + li) = make_float4(0.f, 0.f, 0.f, 0.f);
    }
  }
}

__device__ __forceinline__ float gmag(float gv, float gh) {
  return __builtin_sqrtf(gv * gv + gh * gh + EPSG);
}

// ---------------------------------------------------------------------------
// Kernel 1: per-band partial sums of |grad(input)-grad(target)|
// ---------------------------------------------------------------------------
__global__ __launch_bounds__(BLOCK)
void grad_l1_partial(const float* __restrict__ inp, const float* __restrict__ tgt,
                     float* __restrict__ partial, int bandsPerPlane) {
  __shared__ __align__(16) float sIn[HALO * W];
  __shared__ __align__(16) float sTg[HALO * W];
  __shared__ float wsum[BLOCK / 32];

  const int tid   = threadIdx.x;
  const int plane = blockIdx.x / bandsPerPlane;
  const int band  = blockIdx.x % bandsPerPlane;
  const int r0    = band * TH;
  const float* ip = inp + (size_t)plane * (H * W);
  const float* tp = tgt + (size_t)plane * (H * W);

  stage_tile(ip, sIn, r0, tid);
  stage_tile(tp, sTg, r0, tid);
#if __has_builtin(__builtin_amdgcn_s_wait_asynccnt)
  __builtin_amdgcn_s_wait_asynccnt(0);
#endif
  __syncthreads();

  float acc = 0.f;
#pragma unroll
  for (int i = 0; i < (TH * (W / 4)) / BLOCK; ++i) {     // 8*128/256 = 4
    int g   = i * BLOCK + tid;
    int row = g >> 7;
    int c4  = (g & 127) << 2;
    int up  = row * W + c4;
    int mid = (row + 1) * W + c4;
    int dn  = (row + 2) * W + c4;

    float4 iu = *(const float4*)(sIn + up);
    float4 im = *(const float4*)(sIn + mid);
    float4 id = *(const float4*)(sIn + dn);
    float  il = (c4 > 0)     ? sIn[mid - 1] : 0.f;
    float  ir = (c4 + 4 < W) ? sIn[mid + 4] : 0.f;

    float4 tu = *(const float4*)(sTg + up);
    float4 tm = *(const float4*)(sTg + mid);
    float4 td = *(const float4*)(sTg + dn);
    float  tl = (c4 > 0)     ? sTg[mid - 1] : 0.f;
    float  tr = (c4 + 4 < W) ? sTg[mid + 4] : 0.f;

    acc += __builtin_fabsf(gmag(id.x - iu.x, im.y - il)   - gmag(td.x - tu.x, tm.y - tl));
    acc += __builtin_fabsf(gmag(id.y - iu.y, im.z - im.x) - gmag(td.y - tu.y, tm.z - tm.x));
    acc += __builtin_fabsf(gmag(id.z - iu.z, im.w - im.y) - gmag(td.z - tu.z, tm.w - tm.y));
    acc += __builtin_fabsf(gmag(id.w - iu.w, ir - im.z)   - gmag(td.w - tu.w, tr - tm.z));
  }

  // wave32 reduction
  for (int o = 16; o > 0; o >>= 1) acc += __shfl_down(acc, o, 32);
  if ((tid & 31) == 0) wsum[tid >> 5] = acc;
  __syncthreads();
  if (tid < (BLOCK / 32)) {
    float v = wsum[tid];
    for (int o = 4; o > 0; o >>= 1) v += __shfl_down(v, o, 32);
    if (tid == 0) partial[blockIdx.x] = v;
  }
}

// ---------------------------------------------------------------------------
// Kernel 2: single-wave final reduction of block partials using
// v_wmma_f32_16x16x4_f32 with A = ones(16x4): C accumulates column sums of
// each 4x16 B tile, so sum(C) == 16 * sum(partials) exactly in f32.
// ---------------------------------------------------------------------------
__global__ __launch_bounds__(32)
void grad_l1_reduce(const float* __restrict__ partial, float* __restrict__ out,
                    int n, float scale) {
  int lane = threadIdx.x;
  float s;
#if __has_builtin(__builtin_amdgcn_wmma_f32_16x16x4_f32)
  v2f a; a[0] = 1.0f; a[1] = 1.0f;
  v8f c = {};
  int iters = n >> 6;                    // n is a multiple of 64
  for (int it = 0; it < iters; ++it) {
    const float* p = partial + (it << 6);
    v2f b; b[0] = p[lane]; b[1] = p[lane + 32];
    c = __builtin_amdgcn_wmma_f32_16x16x4_f32(
        /*neg_a=*/false, a, /*neg_b=*/false, b,
        /*c_mod=*/(short)0, c, /*reuse_a=*/false, /*reuse_b=*/false);
  }
  s = (c[0] + c[1] + c[2] + c[3] + c[4] + c[5] + c[6] + c[7]) * 0.0625f;
#else
  s = 0.f;
  for (int i = lane; i < n; i += 32) s += partial[i];
#endif
  for (int o = 16; o > 0; o >>= 1) s += __shfl_down(s, o, 32);
  if (lane == 0) out[0] = s * scale;
}

// ---------------------------------------------------------------------------
extern "C" void kernel_launch(void* const* d_in, const int* in_sizes, int n_in,
                              void* d_out, int out_size, void* d_ws, size_t ws_size,
                              hipStream_t stream) {
  (void)n_in; (void)out_size; (void)ws_size;
  const float* inp = (const float*)d_in[0];
  const float* tgt = (const float*)d_in[1];
  float* out      = (float*)d_out;
  float* partials = (float*)d_ws;

  const long long total = (long long)in_sizes[0];        // 32*3*512*512
  const int planes  = (int)(total / ((long long)H * W)); // 96
  const int bands   = H / TH;                            // 64
  const int nblocks = planes * bands;                    // 6144 (multiple of 64)

  grad_l1_partial<<<nblocks, BLOCK, 0, stream>>>(inp, tgt, partials, bands);

  const float scale = (float)(1.0 / (double)total);      // mean over all elements
  grad_l1_reduce<<<1, 32, 0, stream>>>(partials, out, nblocks, scale);
}